// SoftMoE_40355512714067
// MI455X (gfx1250) — compile-verified
//
#include <hip/hip_runtime.h>
#include <math.h>

#define D_MODEL 1024
#define D_FF    4096
#define NE      8
#define NS      2
#define NSLOT   16      // NE*NS
#define BBATCH  4
#define TT      4096
#define LN_EPS  1e-5f

typedef __attribute__((ext_vector_type(2))) float v2f;
typedef __attribute__((ext_vector_type(8))) float v8f;

__device__ __forceinline__ v8f wmma_f32(v2f a, v2f b, v8f c) {
  // 8 args: (neg_a, A, neg_b, B, c_mod, C, reuse_a, reuse_b)
  return __builtin_amdgcn_wmma_f32_16x16x4_f32(false, a, false, b, (short)0, c,
                                               false, false);
}

// ---------------------------------------------------------------------------
// 1) logits[b,t,n] = (x[b,t,:] . slot_embeds[n,:]) / sqrt(D)
//    One wave per 16-token tile. A = x (M=t, K=d), B = SE^T (K=d, N=n).
// ---------------------------------------------------------------------------
__global__ __launch_bounds__(128) void k_logits(const float* __restrict__ x,
                                                const float* __restrict__ se,
                                                float* __restrict__ logits) {
  const int wave = blockIdx.x * (blockDim.x >> 5) + (threadIdx.x >> 5);
  const int lane = threadIdx.x & 31;
  const int b  = wave / (TT / 16);
  const int t0 = (wave % (TT / 16)) * 16;
  const int mr = lane & 15, g = lane >> 4;
  const float* ap = x + ((size_t)b * TT + t0 + mr) * D_MODEL + 2 * g;
  const float* bp = se + (size_t)mr * D_MODEL + 2 * g;
  v8f acc = {};
#pragma unroll 4
  for (int k = 0; k < D_MODEL; k += 4) {
    v2f a  = *(const v2f*)(ap + k);
    v2f bv = *(const v2f*)(bp + k);
    acc = wmma_f32(a, bv, acc);
  }
#pragma unroll
  for (int i = 0; i < 8; ++i) {
    const int m = i + 8 * g;
    logits[((size_t)b * TT + t0 + m) * NSLOT + mr] = acc[i] * 0.03125f; // 1/sqrt(1024)
  }
}

// ---------------------------------------------------------------------------
// 2) per-(b,n) column stats over tokens for the dispatch softmax
// ---------------------------------------------------------------------------
__global__ __launch_bounds__(256) void k_colstats(const float* __restrict__ logits,
                                                  float* __restrict__ dmax,
                                                  float* __restrict__ dsum) {
  const int b = blockIdx.x >> 4, n = blockIdx.x & 15;
  __shared__ float red[256];
  const int tid = threadIdx.x;
  float m = -3.4e38f;
  for (int t = tid; t < TT; t += 256)
    m = fmaxf(m, logits[((size_t)b * TT + t) * NSLOT + n]);
  red[tid] = m; __syncthreads();
  for (int st = 128; st > 0; st >>= 1) {
    if (tid < st) red[tid] = fmaxf(red[tid], red[tid + st]);
    __syncthreads();
  }
  const float mx = red[0]; __syncthreads();
  float s = 0.f;
  for (int t = tid; t < TT; t += 256)
    s += __expf(logits[((size_t)b * TT + t) * NSLOT + n] - mx);
  red[tid] = s; __syncthreads();
  for (int st = 128; st > 0; st >>= 1) {
    if (tid < st) red[tid] += red[tid + st];
    __syncthreads();
  }
  if (tid == 0) { dmax[blockIdx.x] = mx; dsum[blockIdx.x] = red[0]; }
}

// ---------------------------------------------------------------------------
// 3) slot_in[b,n,d] = sum_t softmax_t(logits)[b,t,n] * x[b,t,d]
//    WMMA with A[m=n,k=t] = exp(logit - max) computed on the fly; /sum at store.
// ---------------------------------------------------------------------------
__global__ __launch_bounds__(128) void k_slotin(const float* __restrict__ logits,
                                                const float* __restrict__ x,
                                                const float* __restrict__ dmax,
                                                const float* __restrict__ dsum,
                                                float* __restrict__ slot_in) {
  const int wave = blockIdx.x * (blockDim.x >> 5) + (threadIdx.x >> 5);
  const int lane = threadIdx.x & 31;
  const int b  = wave / (D_MODEL / 16);
  const int d0 = (wave % (D_MODEL / 16)) * 16;
  const int mr = lane & 15, g = lane >> 4;
  const float mx = dmax[b * NSLOT + mr];
  const float* lp = logits + (size_t)b * TT * NSLOT + mr;
  const float* xp = x + (size_t)b * TT * D_MODEL + d0 + mr;
  v8f acc = {};
  for (int t = 0; t < TT; t += 4) {
    const int k0 = t + 2 * g;
    v2f a, bv;
    a.x  = __expf(lp[(size_t)k0 * NSLOT] - mx);
    a.y  = __expf(lp[(size_t)(k0 + 1) * NSLOT] - mx);
    bv.x = xp[(size_t)k0 * D_MODEL];
    bv.y = xp[(size_t)(k0 + 1) * D_MODEL];
    acc = wmma_f32(a, bv, acc);
  }
#pragma unroll
  for (int i = 0; i < 8; ++i) {
    const int m = i + 8 * g;
    slot_in[((size_t)b * NSLOT + m) * D_MODEL + d0 + mr] =
        acc[i] / dsum[b * NSLOT + m];
  }
}

// ---------------------------------------------------------------------------
// 4) LayerNorm over D per slot row
// ---------------------------------------------------------------------------
__global__ __launch_bounds__(256) void k_ln(const float* __restrict__ slot_in,
                                            const float* __restrict__ gamma,
                                            const float* __restrict__ beta,
                                            float* __restrict__ h) {
  const int row = blockIdx.x;
  const float* p = slot_in + (size_t)row * D_MODEL;
  __shared__ float red[256];
  const int tid = threadIdx.x;
  float s = 0.f;
  for (int i = tid; i < D_MODEL; i += 256) s += p[i];
  red[tid] = s; __syncthreads();
  for (int st = 128; st > 0; st >>= 1) {
    if (tid < st) red[tid] += red[tid + st];
    __syncthreads();
  }
  const float mu = red[0] * (1.0f / D_MODEL); __syncthreads();
  float v = 0.f;
  for (int i = tid; i < D_MODEL; i += 256) { float d = p[i] - mu; v += d * d; }
  red[tid] = v; __syncthreads();
  for (int st = 128; st > 0; st >>= 1) {
    if (tid < st) red[tid] += red[tid + st];
    __syncthreads();
  }
  const float rstd = rsqrtf(red[0] * (1.0f / D_MODEL) + LN_EPS);
  for (int i = tid; i < D_MODEL; i += 256)
    h[(size_t)row * D_MODEL + i] = (p[i] - mu) * rstd * gamma[i] + beta[i];
}

// ---------------------------------------------------------------------------
// 5) u[b,e,s,f] = gelu(h[b,e,s,:] . W1[e,:,f]); M = B*S = 8 rows padded to 16
// ---------------------------------------------------------------------------
__global__ __launch_bounds__(128) void k_mlp1(const float* __restrict__ h,
                                              const float* __restrict__ W1,
                                              float* __restrict__ u) {
  const int wave = blockIdx.x * (blockDim.x >> 5) + (threadIdx.x >> 5);
  const int lane = threadIdx.x & 31;
  const int e  = wave / (D_FF / 16);
  const int f0 = (wave % (D_FF / 16)) * 16;
  const int mr = lane & 15, g = lane >> 4;
  const float* arow = nullptr;
  if (mr < BBATCH * NS) {
    const int bb = mr >> 1, s = mr & 1;
    arow = h + ((size_t)bb * NSLOT + e * NS + s) * D_MODEL;
  }
  const float* bp = W1 + (size_t)e * D_MODEL * D_FF + f0 + mr;
  v8f acc = {};
  for (int k = 0; k < D_MODEL; k += 4) {
    const int k0 = k + 2 * g;
    v2f a, bv;
    a.x  = arow ? arow[k0] : 0.f;
    a.y  = arow ? arow[k0 + 1] : 0.f;
    bv.x = bp[(size_t)k0 * D_FF];
    bv.y = bp[(size_t)(k0 + 1) * D_FF];
    acc = wmma_f32(a, bv, acc);
  }
  if (g == 0) {
#pragma unroll
    for (int i = 0; i < 8; ++i) {
      const int bb = i >> 1, s = i & 1;
      const float vv = acc[i];
      const float ge = 0.5f * vv * (1.0f + erff(vv * 0.70710678118654752f));
      u[(((size_t)bb * NE + e) * NS + s) * D_FF + f0 + mr] = ge;
    }
  }
}

// ---------------------------------------------------------------------------
// 6) slot_out[b,e,s,d] = u[b,e,s,:] . W2[e,:,d]
// ---------------------------------------------------------------------------
__global__ __launch_bounds__(128) void k_mlp2(const float* __restrict__ u,
                                              const float* __restrict__ W2,
                                              float* __restrict__ slot_out) {
  const int wave = blockIdx.x * (blockDim.x >> 5) + (threadIdx.x >> 5);
  const int lane = threadIdx.x & 31;
  const int e  = wave / (D_MODEL / 16);
  const int d0 = (wave % (D_MODEL / 16)) * 16;
  const int mr = lane & 15, g = lane >> 4;
  const float* arow = nullptr;
  if (mr < BBATCH * NS) {
    const int bb = mr >> 1, s = mr & 1;
    arow = u + (((size_t)bb * NE + e) * NS + s) * D_FF;
  }
  const float* bp = W2 + (size_t)e * D_FF * D_MODEL + d0 + mr;
  v8f acc = {};
  for (int k = 0; k < D_FF; k += 4) {
    const int k0 = k + 2 * g;
    v2f a, bv;
    a.x  = arow ? arow[k0] : 0.f;
    a.y  = arow ? arow[k0 + 1] : 0.f;
    bv.x = bp[(size_t)k0 * D_MODEL];
    bv.y = bp[(size_t)(k0 + 1) * D_MODEL];
    acc = wmma_f32(a, bv, acc);
  }
  if (g == 0) {
#pragma unroll
    for (int i = 0; i < 8; ++i) {
      const int bb = i >> 1, s = i & 1;
      slot_out[((size_t)bb * NSLOT + e * NS + s) * D_MODEL + d0 + mr] = acc[i];
    }
  }
}

// ---------------------------------------------------------------------------
// 7) combine[b,t,n] = softmax over the 16 slots (per token)
// ---------------------------------------------------------------------------
__global__ __launch_bounds__(256) void k_combine(const float* __restrict__ logits,
                                                 float* __restrict__ combine) {
  const int tok = blockIdx.x * blockDim.x + threadIdx.x;
  if (tok >= BBATCH * TT) return;
  const float* lp = logits + (size_t)tok * NSLOT;
  float mx = lp[0];
#pragma unroll
  for (int n = 1; n < NSLOT; ++n) mx = fmaxf(mx, lp[n]);
  float s = 0.f, e[NSLOT];
#pragma unroll
  for (int n = 0; n < NSLOT; ++n) { e[n] = __expf(lp[n] - mx); s += e[n]; }
  const float inv = 1.0f / s;
#pragma unroll
  for (int n = 0; n < NSLOT; ++n) combine[(size_t)tok * NSLOT + n] = e[n] * inv;
}

// ---------------------------------------------------------------------------
// 8) out[b,t,d] = combine[b,t,:] . slot_out[b,:,d]; K=16, 4 d-tiles per wave
// ---------------------------------------------------------------------------
__global__ __launch_bounds__(128) void k_out(const float* __restrict__ combine,
                                             const float* __restrict__ slot_out,
                                             float* __restrict__ out) {
  const int wave = blockIdx.x * (blockDim.x >> 5) + (threadIdx.x >> 5);
  const int lane = threadIdx.x & 31;
  const int tilesD = D_MODEL / 64; // 16
  const int b  = wave / ((TT / 16) * tilesD);
  const int r  = wave % ((TT / 16) * tilesD);
  const int t0 = (r / tilesD) * 16;
  const int d0 = (r % tilesD) * 64;
  const int mr = lane & 15, g = lane >> 4;
  const float* cp = combine + ((size_t)b * TT + t0 + mr) * NSLOT + 2 * g;
  v8f acc0 = {}, acc1 = {}, acc2 = {}, acc3 = {};
#pragma unroll
  for (int k = 0; k < NSLOT; k += 4) {
    v2f a = *(const v2f*)(cp + k);
    const float* sp = slot_out + ((size_t)b * NSLOT + k + 2 * g) * D_MODEL + d0 + mr;
    v2f b0, b1, b2, b3;
    b0.x = sp[0];  b0.y = sp[D_MODEL];
    b1.x = sp[16]; b1.y = sp[D_MODEL + 16];
    b2.x = sp[32]; b2.y = sp[D_MODEL + 32];
    b3.x = sp[48]; b3.y = sp[D_MODEL + 48];
    acc0 = wmma_f32(a, b0, acc0);
    acc1 = wmma_f32(a, b1, acc1);
    acc2 = wmma_f32(a, b2, acc2);
    acc3 = wmma_f32(a, b3, acc3);
  }
#pragma unroll
  for (int i = 0; i < 8; ++i) {
    const int m = i + 8 * g;
    float* op = out + ((size_t)b * TT + t0 + m) * D_MODEL + d0 + mr;
    op[0] = acc0[i]; op[16] = acc1[i]; op[32] = acc2[i]; op[48] = acc3[i];
  }
}

// ---------------------------------------------------------------------------
extern "C" void kernel_launch(void* const* d_in, const int* in_sizes, int n_in,
                              void* d_out, int out_size, void* d_ws, size_t ws_size,
                              hipStream_t stream) {
  const float* x     = (const float*)d_in[0];
  const float* se    = (const float*)d_in[1];
  const float* W1    = (const float*)d_in[2];
  const float* W2    = (const float*)d_in[3];
  const float* gamma = (const float*)d_in[4];
  const float* beta  = (const float*)d_in[5];
  float* out = (float*)d_out;

  float* ws       = (float*)d_ws;
  float* logits   = ws;                                     // B*T*16
  float* combine  = logits  + (size_t)BBATCH * TT * NSLOT;  // B*T*16
  float* dmax     = combine + (size_t)BBATCH * TT * NSLOT;  // B*16
  float* dsum     = dmax    + BBATCH * NSLOT;               // B*16
  float* slot_in  = dsum    + BBATCH * NSLOT;               // B*16*D
  float* hbuf     = slot_in + (size_t)BBATCH * NSLOT * D_MODEL;
  float* ubuf     = hbuf    + (size_t)BBATCH * NSLOT * D_MODEL;   // B*E*S*F
  float* slot_out = ubuf    + (size_t)BBATCH * NE * NS * D_FF;    // B*16*D

  k_logits  <<<BBATCH * (TT / 16) / 4,               128, 0, stream>>>(x, se, logits);
  k_colstats<<<BBATCH * NSLOT,                       256, 0, stream>>>(logits, dmax, dsum);
  k_slotin  <<<BBATCH * (D_MODEL / 16) / 4,          128, 0, stream>>>(logits, x, dmax, dsum, slot_in);
  k_ln      <<<BBATCH * NSLOT,                       256, 0, stream>>>(slot_in, gamma, beta, hbuf);
  k_mlp1    <<<NE * (D_FF / 16) / 4,                 128, 0, stream>>>(hbuf, W1, ubuf);
  k_mlp2    <<<NE * (D_MODEL / 16) / 4,              128, 0, stream>>>(ubuf, W2, slot_out);
  k_combine <<<(BBATCH * TT + 255) / 256,            256, 0, stream>>>(logits, combine);
  k_out     <<<BBATCH * (TT / 16) * (D_MODEL / 64) / 4, 128, 0, stream>>>(combine, slot_out, out);
}